// EncModule_83777632076339
// MI455X (gfx1250) — compile-verified
//
#include <hip/hip_runtime.h>
#include <hip/hip_bf16.h>
#include <math.h>

// ---------------------------------------------------------------------------
// EncModule for MI455X (gfx1250, wave32).
// All three GEMMs run on V_WMMA_F32_16X16X4_F32 (fp32 matches reference).
// Async global->LDS staging (ASYNCcnt path) used when the builtin exists.
// Shapes assumed: C=256 (C%128==0), K=32, groups=4, N%64==0.
// ---------------------------------------------------------------------------

#define EPS 1e-5f
#define SLOPE 0.01f
#define KMAX 32

typedef float v2f __attribute__((ext_vector_type(2)));
typedef float v8f __attribute__((ext_vector_type(8)));

#if defined(__has_builtin)
#if __has_builtin(__builtin_amdgcn_global_load_async_to_lds_b128) && \
    __has_builtin(__builtin_amdgcn_s_wait_asynccnt)
#define HAVE_ASYNC_LDS 1
#endif
#endif

#if defined(HAVE_ASYNC_LDS)
typedef int v4i __attribute__((vector_size(16)));
typedef __attribute__((address_space(1))) v4i* gv4i_p;  // global int4*
typedef __attribute__((address_space(3))) v4i* lv4i_p;  // LDS int4*
#endif

__device__ __forceinline__ void copy16_to_lds(const float* gsrc, float* ldst) {
#if defined(HAVE_ASYNC_LDS)
  __builtin_amdgcn_global_load_async_to_lds_b128((gv4i_p)gsrc, (lv4i_p)ldst, 0,
                                                 0);
#else
  *(float4*)ldst = *(const float4*)gsrc;
#endif
}

__device__ __forceinline__ void async_wait() {
#if defined(HAVE_ASYNC_LDS)
  __builtin_amdgcn_s_wait_asynccnt(0);
#endif
}

__device__ __forceinline__ v8f wmma_f32(v2f a, v2f b, v8f c) {
  // D = A(16x4) * B(4x16) + C(16x16), fp32, wave32
  return __builtin_amdgcn_wmma_f32_16x16x4_f32(false, a, false, b,
                                               (short)0, c, false, false);
}

__device__ __forceinline__ float lrelu(float x) {
  return x >= 0.f ? x : SLOPE * x;
}

// ---------------------------------------------------------------------------
// K0: zero the small accumulator region of the workspace
// ---------------------------------------------------------------------------
__global__ void zero_kernel(float* __restrict__ p, int n) {
  int i = blockIdx.x * blockDim.x + threadIdx.x;
  if (i < n) p[i] = 0.f;
}

// ---------------------------------------------------------------------------
// K1: H[b,o,n] = sum_c W[o,c]*X[b,c,n] + bias[o]
//     128x64 workgroup tile, K-chunk 64; each wave owns a 16x64 strip so one
//     A ds_load_b64 feeds 4 WMMAs; X tile stored transposed so B fragments
//     are contiguous ds_load_b64 as well.
//     + per-wave GN1 sum/sumsq partials via wave32 shuffle + atomics.
// ---------------------------------------------------------------------------
__global__ __launch_bounds__(256) void conv_gemm_kernel(
    const float* __restrict__ X, const float* __restrict__ Wm,
    const float* __restrict__ bias, float* __restrict__ H,
    float* __restrict__ gstats, int C, int N) {
  __shared__ float sW[128][64];   // W[o][c] tile (32 KB)
  __shared__ float sXT[64][64];   // X^T[n][c] tile (16 KB)

  const int tid  = threadIdx.x;
  const int lane = tid & 31;
  const int wave = tid >> 5;
  const int n0 = blockIdx.x * 64;
  const int o0 = blockIdx.y * 128;
  const int b  = blockIdx.z;

  const float* Xb = X + (size_t)b * C * N;
  float*       Hb = H + (size_t)b * C * N;

  v8f acc[4];
#pragma unroll
  for (int i = 0; i < 4; ++i)
#pragma unroll
    for (int r = 0; r < 8; ++r) acc[i][r] = 0.f;

  const int m  = lane & 15;        // A: row / B,D: column within tile
  const int kb = (lane >> 4) * 2;  // K pair selected by half-wave
  const int tm = wave * 16;        // wave's M strip inside the 128 rows

  for (int c0 = 0; c0 < C; c0 += 64) {
    // stage W tile: 2048 float4, async global->LDS when available
#pragma unroll
    for (int i = 0; i < 8; ++i) {
      int j = tid + i * 256;
      int row = j >> 4, col = (j & 15) * 4;
      copy16_to_lds(&Wm[(size_t)(o0 + row) * C + c0 + col], &sW[row][col]);
    }
    // stage X tile transposed (coalesced float4 read over n)
#pragma unroll
    for (int i = 0; i < 4; ++i) {
      int j = tid + i * 256;
      int cc = j >> 4, nc = (j & 15) * 4;
      float4 v = *(const float4*)&Xb[(size_t)(c0 + cc) * N + n0 + nc];
      sXT[nc + 0][cc] = v.x;
      sXT[nc + 1][cc] = v.y;
      sXT[nc + 2][cc] = v.z;
      sXT[nc + 3][cc] = v.w;
    }
    async_wait();
    __syncthreads();

#pragma unroll 2
    for (int kk = 0; kk < 64; kk += 4) {
      v2f a = *(const v2f*)&sW[tm + m][kk + kb];
#pragma unroll
      for (int i = 0; i < 4; ++i) {
        v2f bv = *(const v2f*)&sXT[i * 16 + m][kk + kb];
        acc[i] = wmma_f32(a, bv, acc[i]);
      }
    }
    __syncthreads();
  }

  // epilogue: +bias, store, accumulate GN1 partials (wave strip is inside
  // one group: group size = C/4 = 64 rows, strip = 16 rows, 16-aligned)
  float s = 0.f, s2 = 0.f;
#pragma unroll
  for (int i = 0; i < 4; ++i) {
#pragma unroll
    for (int r = 0; r < 8; ++r) {
      int ml = tm + r + ((lane >> 4) << 3);
      int o  = o0 + ml;
      int n  = n0 + i * 16 + m;
      float v = acc[i][r] + bias[o];
      Hb[(size_t)o * N + n] = v;
      s += v;
      s2 += v * v;
    }
  }
#pragma unroll
  for (int off = 16; off > 0; off >>= 1) {
    s  += __shfl_down(s, off);
    s2 += __shfl_down(s2, off);
  }
  if (lane == 0) {
    int g = (o0 + tm) / (C >> 2);
    atomicAdd(&gstats[(b * 4 + g) * 2 + 0], s);
    atomicAdd(&gstats[(b * 4 + g) * 2 + 1], s2);
  }
}

// ---------------------------------------------------------------------------
// K2: finalize GN1 stats -> (mean, rsqrt(var+eps)); c2[k] = sum_c cw[k,c]^2
// ---------------------------------------------------------------------------
__global__ void finalize_stats_kernel(float* __restrict__ gstats,
                                      const float* __restrict__ cw,
                                      float* __restrict__ c2,
                                      int B, int C, int N, int K) {
  int t = threadIdx.x;
  if (t < B * 4) {
    float cnt  = (float)(C >> 2) * (float)N;
    float mean = gstats[t * 2] / cnt;
    float var  = gstats[t * 2 + 1] / cnt - mean * mean;
    gstats[t * 2]     = mean;
    gstats[t * 2 + 1] = rsqrtf(var + EPS);
  }
  if (t >= 32 && (t - 32) < K) {
    int k = t - 32;
    float s = 0.f;
    for (int c = 0; c < C; ++c) {
      float v = cw[(size_t)k * C + c];
      s += v * v;
    }
    c2[k] = s;
  }
}

// ---------------------------------------------------------------------------
// K3: fused GN1-normalize + leaky + xc GEMM + softmax + E^T accumulation
//     one workgroup per 64 spatial positions
// ---------------------------------------------------------------------------
__global__ __launch_bounds__(256) void encode_kernel(
    const float* __restrict__ H, const float* __restrict__ cw,
    const float* __restrict__ scale, const float* __restrict__ c2,
    const float* __restrict__ gn1w, const float* __restrict__ gn1b,
    const float* __restrict__ gstats,
    float* __restrict__ Ebuf, float* __restrict__ Asum,
    int C, int N, int K) {
  __shared__ float sxf[64][256];   // xf[n][c]      (64 KB)
  __shared__ float scw[KMAX][256]; // codewords     (32 KB)
  __shared__ float sxc[64][KMAX];  // xc then A     (8 KB)
  __shared__ float sx2[64];
  __shared__ float smean[4], sistd[4];

  const int tid  = threadIdx.x;
  const int lane = tid & 31;
  const int wave = tid >> 5;
  const int n0 = blockIdx.x * 64;
  const int b  = blockIdx.y;
  const float* Hb = H + (size_t)b * C * N;

  if (tid < 4) {
    smean[tid] = gstats[(b * 4 + tid) * 2];
    sistd[tid] = gstats[(b * 4 + tid) * 2 + 1];
  }
  // stage codewords (row-major contiguous, async when available)
  {
    float* scwf = &scw[0][0];
    for (int j = tid; j < (K * C) / 4; j += 256)
      copy16_to_lds(&cw[(size_t)j * 4], &scwf[j * 4]);
  }
  __syncthreads();

  // stage xf = leaky(GN1(h)) (coalesced over n)
  const int cpg = C >> 2;
  for (int idx = tid; idx < C * 64; idx += 256) {
    int nl = idx & 63, c = idx >> 6;
    int g = c / cpg;
    float v  = Hb[(size_t)c * N + n0 + nl];
    float xn = (v - smean[g]) * sistd[g] * gn1w[c] + gn1b[c];
    sxf[nl][c] = lrelu(xn);
  }
  async_wait();
  __syncthreads();

  // x2[n]
  if (tid < 64) {
    float s = 0.f;
    for (int c = 0; c < C; ++c) {
      float v = sxf[tid][c];
      s += v * v;
    }
    sx2[tid] = s;
  }
  __syncthreads();

  // xc(64x32) = xf(64xC) @ cw^T(Cx32) : 8 tiles, one per wave
  {
    const int m = lane & 15, kb = (lane >> 4) * 2;
    const int tm = (wave >> 1) * 16;  // n tile
    const int tn = (wave & 1) * 16;   // k tile
    v8f acc;
#pragma unroll
    for (int r = 0; r < 8; ++r) acc[r] = 0.f;
    for (int cc = 0; cc < C; cc += 4) {
      v2f a  = *(const v2f*)&sxf[tm + m][cc + kb];
      v2f bv = *(const v2f*)&scw[tn + m][cc + kb];  // B[c][k] = cw[k][c]
      acc = wmma_f32(a, bv, acc);
    }
#pragma unroll
    for (int r = 0; r < 8; ++r) {
      int nr = tm + r + ((lane >> 4) << 3);
      sxc[nr][tn + m] = acc[r];
    }
  }
  __syncthreads();

  // softmax over K per position, A written back into sxc
  if (tid < 64) {
    float x2v = sx2[tid];
    float sl[KMAX];
    float mx = -1e30f;
    for (int k = 0; k < K; ++k) {
      float v = scale[k] * (x2v + c2[k] - 2.f * sxc[tid][k]);
      sl[k] = v;
      mx = fmaxf(mx, v);
    }
    float ss = 0.f;
    for (int k = 0; k < K; ++k) {
      float e = __expf(sl[k] - mx);
      sl[k] = e;
      ss += e;
    }
    float inv = 1.f / ss;
    for (int k = 0; k < K; ++k) sxc[tid][k] = sl[k] * inv;
  }
  __syncthreads();

  // Asum[b][k] += sum_n A
  if (tid < K) {
    float s = 0.f;
    for (int n = 0; n < 64; ++n) s += sxc[n][tid];
    atomicAdd(&Asum[b * K + tid], s);
  }

  // E(32xC) += A^T(32x64) @ xf(64xC) : 32 tiles, 4 per wave
  {
    const int m = lane & 15, kb = (lane >> 4) * 2;
#pragma unroll
    for (int i = 0; i < 4; ++i) {
      int tile = wave * 4 + i;
      int tk = (tile >> 4) * 16;  // k tile
      int tc = (tile & 15) * 16;  // c tile
      v8f acc;
#pragma unroll
      for (int r = 0; r < 8; ++r) acc[r] = 0.f;
#pragma unroll 4
      for (int nn = 0; nn < 64; nn += 4) {
        v2f a, bv;
        a.x = sxc[nn + kb][tk + m];  // A^T[k][n] = A[n][k]
        a.y = sxc[nn + kb + 1][tk + m];
        bv.x = sxf[nn + kb][tc + m];
        bv.y = sxf[nn + kb + 1][tc + m];
        acc = wmma_f32(a, bv, acc);
      }
#pragma unroll
      for (int r = 0; r < 8; ++r) {
        int kr = tk + r + ((lane >> 4) << 3);
        atomicAdd(&Ebuf[((size_t)b * K + kr) * C + tc + m], acc[r]);
      }
    }
  }
}

// ---------------------------------------------------------------------------
// K4: E -= Asum*cw ; GN2 + leaky ; en = mean_k ; gamma = sigmoid(fc) ; se
// ---------------------------------------------------------------------------
__global__ __launch_bounds__(256) void head_kernel(
    const float* __restrict__ Ebuf, const float* __restrict__ Asum,
    const float* __restrict__ cw,
    const float* __restrict__ gn2w, const float* __restrict__ gn2b,
    const float* __restrict__ fcw, const float* __restrict__ fcb,
    const float* __restrict__ sew, const float* __restrict__ seb,
    float* __restrict__ out_en, float* __restrict__ out_se,
    float* __restrict__ gamma, int B, int C, int K, int NC) {
  __shared__ float sE[KMAX][256];
  __shared__ float sen[256];
  __shared__ float sstat[8];

  const int tid = threadIdx.x;
  const int kpg = K >> 2;

  for (int b = 0; b < B; ++b) {
    for (int idx = tid; idx < K * C; idx += 256) {
      int k = idx / C, c = idx % C;
      sE[k][c] = Ebuf[((size_t)b * K + k) * C + c] -
                 Asum[b * K + k] * cw[(size_t)k * C + c];
    }
    __syncthreads();
    if (tid < 4) {  // per-group stats over kpg*C elements (tiny)
      float s = 0.f, s2 = 0.f;
      for (int k = tid * kpg; k < (tid + 1) * kpg; ++k)
        for (int c = 0; c < C; ++c) {
          float v = sE[k][c];
          s += v;
          s2 += v * v;
        }
      float cnt = (float)(kpg * C);
      float mean = s / cnt;
      float var = s2 / cnt - mean * mean;
      sstat[tid * 2] = mean;
      sstat[tid * 2 + 1] = rsqrtf(var + EPS);
    }
    __syncthreads();
    for (int idx = tid; idx < K * C; idx += 256) {
      int k = idx / C, c = idx % C;
      int g = k / kpg;
      float v = (sE[k][c] - sstat[g * 2]) * sstat[g * 2 + 1] * gn2w[k] + gn2b[k];
      sE[k][c] = lrelu(v);
    }
    __syncthreads();
    if (tid < C) {
      float s = 0.f;
      for (int k = 0; k < K; ++k) s += sE[k][tid];
      float e = s / (float)K;
      sen[tid] = e;
      out_en[(size_t)b * C + tid] = e;
    }
    __syncthreads();
    if (tid < C) {
      float s = fcb[tid];
      for (int c = 0; c < C; ++c) s += sen[c] * fcw[(size_t)tid * C + c];
      gamma[(size_t)b * C + tid] = 1.f / (1.f + __expf(-s));
    }
    if (tid < NC) {
      float s = seb[tid];
      for (int c = 0; c < C; ++c) s += sen[c] * sew[(size_t)tid * C + c];
      out_se[b * NC + tid] = s;
    }
    __syncthreads();
  }
}

// ---------------------------------------------------------------------------
// K5: out = relu(x * (1 + gamma[b,c])), float4 vectorized.
//     grid = (N/1024, B*C): channel index comes from blockIdx.y (no int div).
// ---------------------------------------------------------------------------
__global__ __launch_bounds__(256) void residual_kernel(
    const float* __restrict__ X, const float* __restrict__ gamma,
    float* __restrict__ Out, long long N) {
  const long long cl = blockIdx.y;  // flat b*C + c
  const long long base =
      cl * N + ((long long)blockIdx.x * 256 + threadIdx.x) * 4;
  float mscale = 1.f + gamma[cl];
  float4 xv = *(const float4*)&X[base];
  float4 o;
  o.x = fmaxf(xv.x * mscale, 0.f);
  o.y = fmaxf(xv.y * mscale, 0.f);
  o.z = fmaxf(xv.z * mscale, 0.f);
  o.w = fmaxf(xv.w * mscale, 0.f);
  *(float4*)&Out[base] = o;
}

// ---------------------------------------------------------------------------
extern "C" void kernel_launch(void* const* d_in, const int* in_sizes, int n_in,
                              void* d_out, int out_size, void* d_ws,
                              size_t ws_size, hipStream_t stream) {
  const float* x     = (const float*)d_in[0];
  const float* convw = (const float*)d_in[1];
  const float* convb = (const float*)d_in[2];
  const float* gn1w  = (const float*)d_in[3];
  const float* gn1b  = (const float*)d_in[4];
  const float* cw    = (const float*)d_in[5];
  const float* scl   = (const float*)d_in[6];
  const float* gn2w  = (const float*)d_in[7];
  const float* gn2b  = (const float*)d_in[8];
  const float* fcw   = (const float*)d_in[9];
  const float* fcb   = (const float*)d_in[10];
  const float* sew   = (const float*)d_in[11];
  const float* seb   = (const float*)d_in[12];

  const int C  = in_sizes[2];   // 256
  const int K  = in_sizes[6];   // 32
  const int NC = in_sizes[12];  // 4
  const long long xtot = (long long)in_sizes[0];
  const int B = (int)(((long long)out_size - xtot) / (C + NC));
  const long long N = xtot / ((long long)B * C);

  // workspace layout
  float* Hbuf   = (float*)d_ws;                 // B*C*N
  float* smallb = Hbuf + xtot;
  float* gstats = smallb;                       // B*4*2
  float* Asum   = gstats + (size_t)B * 8;       // B*K
  float* c2     = Asum + (size_t)B * K;         // K
  float* Ebuf   = c2 + K;                       // B*K*C
  float* gamma  = Ebuf + (size_t)B * K * C;     // B*C

  float* out0   = (float*)d_out;
  float* out_en = out0 + xtot;
  float* out_se = out_en + (long long)B * C;

  const int nsmall = B * 8 + B * K + K + B * K * C + B * C;
  zero_kernel<<<(nsmall + 255) / 256, 256, 0, stream>>>(smallb, nsmall);

  dim3 g1((unsigned)(N / 64), (unsigned)(C / 128), (unsigned)B);
  conv_gemm_kernel<<<g1, 256, 0, stream>>>(x, convw, convb, Hbuf, gstats, C,
                                           (int)N);

  finalize_stats_kernel<<<1, 64, 0, stream>>>(gstats, cw, c2, B, C, (int)N, K);

  dim3 g3((unsigned)(N / 64), (unsigned)B);
  encode_kernel<<<g3, 256, 0, stream>>>(Hbuf, cw, scl, c2, gn1w, gn1b, gstats,
                                        Ebuf, Asum, C, (int)N, K);

  head_kernel<<<1, 256, 0, stream>>>(Ebuf, Asum, cw, gn2w, gn2b, fcw, fcb, sew,
                                     seb, out_en, out_se, gamma, B, C, K, NC);

  dim3 g5((unsigned)(N / 1024), (unsigned)(B * C));
  residual_kernel<<<g5, 256, 0, stream>>>(x, gamma, out0, N);
}